// AdaptiveRouterMultiStep_21414706938043
// MI455X (gfx1250) — compile-verified
//
#include <hip/hip_runtime.h>

typedef __attribute__((ext_vector_type(2))) float v2f;
typedef __attribute__((ext_vector_type(8))) float v8f;

#define Bn 16
#define Hh 112
#define Wd 112
#define Cc 128
#define Ll 12544   // Hh*Wd
#define DH 64
#define KK 4

// ---------------------------------------------------------------------------
// K0: loop-invariant precompute: qk[h][c] = (Wk[:,h,:] @ q[h,:]) / sqrt(dh),
// coef = (H*W) * softplus(alpha). Runs once.
// ---------------------------------------------------------------------------
__global__ __launch_bounds__(256) void k_init(const float* __restrict__ Wk,
                                              const float* __restrict__ q,
                                              const float* __restrict__ alpha,
                                              float* __restrict__ qk,
                                              float* __restrict__ coef) {
  int t = threadIdx.x;
  int h = t >> 7, c = t & 127;
  float s = 0.f;
  for (int d = 0; d < DH; ++d) s += Wk[c * 128 + h * 64 + d] * q[h * 64 + d];
  qk[h * 128 + c] = s * 0.125f;  // fold 1/sqrt(64)
  if (t == 0) {
    float a = alpha[0];
    float sp = fmaxf(a, 0.f) + log1pf(__expf(-fabsf(a)));  // stable softplus
    coef[0] = (float)(Hh * Wd) * sp;
  }
}

// ---------------------------------------------------------------------------
// K1: scores[b,h,l] = x[b,l,:] . qk[h,:]  (+ per-64-pixel-tile max partials)
// grid = (196, B), block = 256. 64 pixels/block staged in padded LDS.
// ---------------------------------------------------------------------------
__global__ __launch_bounds__(256) void k_scores(const float* __restrict__ xin,
                                                const float* __restrict__ qk,
                                                float* __restrict__ scores,
                                                float* __restrict__ pmax) {
  __shared__ float xs[64 * 132];   // 64 pixels x 128 ch, padded rows
  __shared__ float red[512];
  __shared__ float sred[128];
  int t = threadIdx.x;
  int b = blockIdx.y, tile = blockIdx.x;
  int l0 = tile * 64;

  const float4* xg = (const float4*)(xin + ((size_t)b * Ll + l0) * Cc);
  for (int j = 0; j < 8; ++j) {
    int f = t + j * 256;         // float4 index 0..2047
    int p = f >> 5;              // pixel 0..63
    int c = (f & 31) << 2;       // channel (mult of 4)
    *(float4*)(&xs[p * 132 + c]) = xg[f];
  }
  __syncthreads();

  int p = t >> 2, part = t & 3;
  const float* row = &xs[p * 132 + part * 32];
  const float* q0 = &qk[part * 32];
  const float* q1 = &qk[128 + part * 32];
  float s0 = 0.f, s1 = 0.f;
  for (int j = 0; j < 32; ++j) { float xv = row[j]; s0 += xv * q0[j]; s1 += xv * q1[j]; }
  red[t] = s0; red[256 + t] = s1;
  __syncthreads();

  if (t < 64) {
    float a0 = red[4 * t] + red[4 * t + 1] + red[4 * t + 2] + red[4 * t + 3];
    float a1 = red[256 + 4 * t] + red[256 + 4 * t + 1] + red[256 + 4 * t + 2] + red[256 + 4 * t + 3];
    scores[((size_t)(b * 2 + 0)) * Ll + l0 + t] = a0;
    scores[((size_t)(b * 2 + 1)) * Ll + l0 + t] = a1;
    sred[t] = a0; sred[64 + t] = a1;
  }
  __syncthreads();
  for (int off = 32; off > 0; off >>= 1) {
    if (t < off) sred[t] = fmaxf(sred[t], sred[t + off]);
    else if (t >= 64 && t < 64 + off) sred[t] = fmaxf(sred[t], sred[t + off]);
    __syncthreads();
  }
  if (t == 0)  pmax[(b * 2 + 0) * 196 + tile] = sred[0];
  if (t == 64) pmax[(b * 2 + 1) * 196 + tile] = sred[64];
}

// ---------------------------------------------------------------------------
// K2: per (b,h): global max m, Z = sum exp(s-m), and
//     xattn[b,h,c] = sum_l softmax(s)[l] * x[b,l,c].   grid = 32, block = 256.
// ---------------------------------------------------------------------------
__global__ __launch_bounds__(256) void k_reduce(const float* __restrict__ xin,
                                                const float* __restrict__ scores,
                                                const float* __restrict__ pmax,
                                                float* __restrict__ marr,
                                                float* __restrict__ izarr,
                                                float* __restrict__ xattn) {
  __shared__ float lds[256];
  __shared__ float mz[2];
  int t = threadIdx.x;
  int bh = blockIdx.x, b = bh >> 1;
  const float* sp = scores + (size_t)bh * Ll;

  float v = -3.4e38f;
  if (t < 196) v = pmax[bh * 196 + t];
  lds[t] = v; __syncthreads();
  for (int off = 128; off > 0; off >>= 1) {
    if (t < off) lds[t] = fmaxf(lds[t], lds[t + off]);
    __syncthreads();
  }
  if (t == 0) mz[0] = lds[0];
  __syncthreads();
  float m = mz[0];

  float z = 0.f;
  for (int l = t; l < Ll; l += 256) z += __expf(sp[l] - m);
  __syncthreads();
  lds[t] = z; __syncthreads();
  for (int off = 128; off > 0; off >>= 1) {
    if (t < off) lds[t] += lds[t + off];
    __syncthreads();
  }
  if (t == 0) { mz[1] = 1.f / lds[0]; marr[bh] = m; izarr[bh] = mz[1]; }
  __syncthreads();
  float invZ = mz[1];

  int c = t & 127, g = t >> 7;
  const float* xb = xin + (size_t)b * Ll * Cc;
  float acc = 0.f;
  int lbeg = g * (Ll / 2), lend = lbeg + Ll / 2;
  for (int l = lbeg; l < lend; ++l)
    acc += __expf(sp[l] - m) * xb[(size_t)l * Cc + c];
  __syncthreads();
  lds[t] = acc; __syncthreads();
  if (t < 128) xattn[bh * Cc + c] = (lds[c] + lds[128 + c]) * invZ;
}

// ---------------------------------------------------------------------------
// K3: per batch: pooled -> logits -> softmax/top2 -> wst ->
//     Mt[d][c] = sum_k wst[k] * Wb[k][c][d]   (TRANSPOSED branch matrix)
// grid = B, block = 256.
// ---------------------------------------------------------------------------
__global__ __launch_bounds__(256) void k_finish(const float* __restrict__ xattn,
                                                const float* __restrict__ Wv,
                                                const float* __restrict__ Wmlp,
                                                const float* __restrict__ Wb,
                                                float* __restrict__ Mtbuf) {
  __shared__ float xa[256];
  __shared__ float pooled[128];
  __shared__ float logits[4];
  __shared__ float wst[4];
  int t = threadIdx.x, b = blockIdx.x;

  xa[t] = xattn[b * 256 + t];
  __syncthreads();
  if (t < 128) {
    int h = t >> 6;
    const float* xh = &xa[h * 128];
    float s = 0.f;
    for (int c = 0; c < 128; ++c) s += xh[c] * Wv[c * 128 + t];  // col = h*64+d = t
    pooled[t] = s;
  }
  __syncthreads();
  if (t < 4) {
    float s = 0.f;
    for (int i = 0; i < 128; ++i) s += pooled[i] * Wmlp[i * 4 + t];
    logits[t] = s;
  }
  __syncthreads();
  if (t == 0) {
    float w[4]; float mx = -3.4e38f;
    for (int k = 0; k < 4; ++k) { w[k] = logits[k] / 1.5f; mx = fmaxf(mx, w[k]); }
    float sum = 0.f;
    for (int k = 0; k < 4; ++k) { w[k] = __expf(w[k] - mx); sum += w[k]; }
    for (int k = 0; k < 4; ++k) w[k] /= sum;
    int i1 = 0;
    for (int k = 1; k < 4; ++k) if (w[k] > w[i1]) i1 = k;     // first max (jax tie rule)
    int i2 = -1;
    for (int k = 0; k < 4; ++k) if (k != i1 && (i2 < 0 || w[k] > w[i2])) i2 = k;
    float inv = 1.f / (w[i1] + w[i2] + 1e-9f);
    for (int k = 0; k < 4; ++k) {
      float wsk = (k == i1 || k == i2) ? w[k] * inv : 0.f;    // renormalized top-n
      wst[k] = wsk + (w[k] - wsk);                            // straight-through fwd
    }
  }
  __syncthreads();
  float w0 = wst[0], w1 = wst[1], w2 = wst[2], w3 = wst[3];
  float* Mo = Mtbuf + (size_t)b * 16384;
  for (int idx = t; idx < 16384; idx += 256) {
    int d = idx >> 7, c = idx & 127;          // Mt[d*128+c] = M[c][d]
    int e = c * 128 + d;
    Mo[idx] = w0 * Wb[e] + w1 * Wb[16384 + e] + w2 * Wb[32768 + e] + w3 * Wb[49152 + e];
  }
}

// ---------------------------------------------------------------------------
// K4: heavy kernel: x_out[b,l,:] = (x[b,l,:] * f(l)) @ M_b  via fp32 WMMA.
// f(l) = 1 + coef * 0.5*(p0(l)+p1(l)) recomputed from scores + (m, 1/Z).
// grid = (49, B), block = 256 (8 waves). 256 pixels/block, 16 subtiles of 16.
// LDS (dynamic): Mt transposed+padded 128x132 (67.6KB) + A tile 16x132 (8.3KB).
// B fragment = contiguous ds_load_b64 (conflict-free: bank = (4*bn+k)%64).
// ---------------------------------------------------------------------------
__global__ __launch_bounds__(256) void k_conv(const float* __restrict__ xin,
                                              float* __restrict__ xout,
                                              const float* __restrict__ scores,
                                              const float* __restrict__ marr,
                                              const float* __restrict__ izarr,
                                              const float* __restrict__ coefp,
                                              const float* __restrict__ Mtbuf) {
  extern __shared__ float sm[];
  float* Ms = sm;            // Mt: 128 rows (n) x 132 (k padded)
  float* As = sm + 128 * 132;  // 16 x 132 (padded)

  int t = threadIdx.x;
  int b = blockIdx.y, tile = blockIdx.x;
  int l0 = tile * 256;

  const float4* Mg = (const float4*)(Mtbuf + (size_t)b * 16384);
  for (int i4 = t; i4 < 4096; i4 += 256) {
    int flat = i4 * 4;
    int d = flat >> 7, c = flat & 127;
    *(float4*)(&Ms[d * 132 + c]) = Mg[i4];   // conflict-free padded store
  }

  float coef = coefp[0];
  float m0 = marr[b * 2], m1 = marr[b * 2 + 1];
  float iz0 = izarr[b * 2], iz1 = izarr[b * 2 + 1];
  const float* s0p = scores + (size_t)(b * 2 + 0) * Ll + l0;
  const float* s1p = scores + (size_t)(b * 2 + 1) * Ll + l0;

  int lane = t & 31, wave = t >> 5;
  int p = t >> 4, c0 = (t & 15) * 8;           // staging: 16 thr/pixel, 8 ch each
  int arow = lane & 15, kof = (lane >> 4) * 2; // A frag: 16x4 f32 layout
  int n0 = wave * 16;
  int bn = n0 + (lane & 15);                   // B frag row (output channel)
  int orow = (lane >> 4) * 8;                  // C/D: VGPR i -> M = i (+8 hi half)
  int ocol = lane & 15;

  const float* arp = &As[arow * 132];
  const float* brp = &Ms[bn * 132];

  for (int st = 0; st < 16; ++st) {
    int lp = st * 16;
    float f = 1.0f + coef * 0.5f * (__expf(s0p[lp + p] - m0) * iz0 +
                                    __expf(s1p[lp + p] - m1) * iz1);
    const float* xr = xin + ((size_t)b * Ll + l0 + lp + p) * Cc + c0;
    float4 a0 = *(const float4*)xr;
    float4 a1 = *(const float4*)(xr + 4);
    a0.x *= f; a0.y *= f; a0.z *= f; a0.w *= f;
    a1.x *= f; a1.y *= f; a1.z *= f; a1.w *= f;
    __syncthreads();                 // previous subtile finished reading As
    float* dst = &As[p * 132 + c0];
    *(float4*)dst = a0;
    *(float4*)(dst + 4) = a1;
    __syncthreads();                 // As (and Ms on first iter) visible

    v8f acc = {0.f, 0.f, 0.f, 0.f, 0.f, 0.f, 0.f, 0.f};
    #pragma unroll
    for (int k0 = 0; k0 < 128; k0 += 4) {
      v2f Af = *(const v2f*)(&arp[k0 + kof]);  // ds_load_b64, contiguous
      v2f Bf = *(const v2f*)(&brp[k0 + kof]);  // ds_load_b64, contiguous
      acc = __builtin_amdgcn_wmma_f32_16x16x4_f32(false, Af, false, Bf,
                                                  (short)0, acc, false, false);
    }
    float* ob = xout + ((size_t)b * Ll + l0 + lp) * Cc + n0 + ocol;
    #pragma unroll
    for (int i = 0; i < 8; ++i) ob[(size_t)(i + orow) * Cc] = acc[i];
  }
}

// ---------------------------------------------------------------------------
extern "C" void kernel_launch(void* const* d_in, const int* in_sizes, int n_in,
                              void* d_out, int out_size, void* d_ws, size_t ws_size,
                              hipStream_t stream) {
  const float* x     = (const float*)d_in[0];
  const float* Wk    = (const float*)d_in[1];
  const float* Wv    = (const float*)d_in[2];
  const float* q     = (const float*)d_in[3];
  const float* Wmlp  = (const float*)d_in[4];
  const float* Wb    = (const float*)d_in[5];
  const float* alpha = (const float*)d_in[6];
  float* out = (float*)d_out;
  float* w = (float*)d_ws;

  const size_t XBUF   = 0;
  const size_t QK     = (size_t)Bn * Ll * Cc;        // 25,690,112 floats
  const size_t COEF   = QK + 256;
  const size_t SCORES = COEF + 16;
  const size_t PMAX   = SCORES + (size_t)Bn * 2 * Ll;
  const size_t MARR   = PMAX + (size_t)Bn * 2 * 196;
  const size_t IZ     = MARR + 32;
  const size_t XATT   = IZ + 32;
  const size_t MBUF   = XATT + (size_t)Bn * 2 * Cc;

  k_init<<<1, 256, 0, stream>>>(Wk, q, alpha, w + QK, w + COEF);

  const size_t conv_lds = (size_t)(128 * 132 + 16 * 132) * sizeof(float); // 76KB
  for (int s = 0; s < 3; ++s) {
    const float* xi = (s == 0) ? x : (w + XBUF);
    float* xo = (s == 2) ? out : (w + XBUF);
    k_scores<<<dim3(196, Bn), 256, 0, stream>>>(xi, w + QK, w + SCORES, w + PMAX);
    k_reduce<<<Bn * 2, 256, 0, stream>>>(xi, w + SCORES, w + PMAX,
                                         w + MARR, w + IZ, w + XATT);
    k_finish<<<Bn, 256, 0, stream>>>(w + XATT, Wv, Wmlp, Wb, w + MBUF);
    k_conv<<<dim3(49, Bn), 256, conv_lds, stream>>>(
        xi, xo, w + SCORES, w + MARR, w + IZ, w + COEF, w + MBUF);
  }
  (void)in_sizes; (void)n_in; (void)out_size; (void)ws_size;
}